// SimGNN_2791728743006
// MI455X (gfx1250) — compile-verified
//
#include <hip/hip_runtime.h>
#include <hip/hip_bf16.h>
#include <math.h>

#define NNODES 50000
#define NEDGES 800000

typedef __attribute__((ext_vector_type(2))) float v2f;
typedef __attribute__((ext_vector_type(8))) float v8f;

// ---------------------------------------------------------------------------
// Y[M x Nd] = X[M x Kd] @ W[Kd x Nd], full fp32 via V_WMMA_F32_16X16X4_F32.
// W is staged in LDS once per workgroup. One wave owns one 16-row m-tile and
// computes ALL n-tiles (NT of them), so each A fragment (global_load_b64) is
// reused NT times; B fragments come from LDS (ds_load).
//   A 16x4 f32: lane l(0..15)->M=l holds K={k,k+1}; lane l+16 holds K={k+2,k+3}.
//   B 4x16 f32: v.x lanes0-15 -> row k, lanes16-31 -> row k+2; v.y -> k+1/k+3.
//   C 16x16 f32: VGPR r, lanes0-15 -> M=r, lanes16-31 -> M=8+r, N=lane&15.
// ---------------------------------------------------------------------------
template <int NT>   // Nd = NT*16 (NT = 4 for F=64, NT = 2 for F=32)
__global__ __launch_bounds__(256)
void gemm_wmma_lds(const float* __restrict__ X, const float* __restrict__ W,
                   float* __restrict__ Y, int M, int Kd) {
  const int Nd = NT * 16;
  __shared__ float Wl[64 * 64];                    // <= 16 KB of the 320 KB WGP LDS
  for (int i = threadIdx.x; i < Kd * Nd; i += blockDim.x) Wl[i] = W[i];
  __syncthreads();                                  // all waves participate first

  int mtile = (blockIdx.x * blockDim.x + threadIdx.x) >> 5;  // wave id == m-tile
  int lane  = threadIdx.x & 31;
  if (mtile * 16 >= M) return;                      // whole-wave exit, EXEC stays full
  const int m0 = mtile * 16;
  const int ml = lane & 15;
  const int hi = lane >> 4;                         // 0 or 1

  v8f c[NT];
#pragma unroll
  for (int t = 0; t < NT; ++t) c[t] = (v8f){};

  const float* xr = X + (size_t)(m0 + ml) * Kd + 2 * hi;
  for (int k = 0; k < Kd; k += 4) {
    v2f a;
    a.x = xr[k];
    a.y = xr[k + 1];                                // contiguous -> global_load_b64
#pragma unroll
    for (int t = 0; t < NT; ++t) {
      v2f b;
      b.x = Wl[(k + 2 * hi) * Nd + t * 16 + ml];
      b.y = Wl[(k + 2 * hi + 1) * Nd + t * 16 + ml];
      c[t] = __builtin_amdgcn_wmma_f32_16x16x4_f32(false, a, false, b, (short)0,
                                                   c[t], false, false);
    }
  }
#pragma unroll
  for (int t = 0; t < NT; ++t) {
    float* yr = Y + (size_t)(m0 + 8 * hi) * Nd + t * 16 + ml;
#pragma unroll
    for (int r = 0; r < 8; ++r) yr[(size_t)r * Nd] = c[t][r];
  }
}

// ---------------------------------------------------------------------------
// Degree / normalization  (self-loop contributes the initial 1.0)
// ---------------------------------------------------------------------------
__global__ void deg_init(float* __restrict__ deg, int n) {
  int t = blockIdx.x * blockDim.x + threadIdx.x;
  if (t < n) deg[t] = 1.0f;
}
__global__ void deg_count(const int* __restrict__ dst, float* __restrict__ deg, int e) {
  int t = blockIdx.x * blockDim.x + threadIdx.x;
  if (t < e) atomicAdd(&deg[dst[t]], 1.0f);
}
__global__ void deg_fin(float* __restrict__ deg, int n) {
  int t = blockIdx.x * blockDim.x + threadIdx.x;
  if (t < n) deg[t] = rsqrtf(deg[t]);              // deg >= 1 always (self-loops)
}

// out[i,f] = H[i,f] * dinv[i]^2  (self-loop term; also zero-inits accumulator)
__global__ void selfloop_init(const float* __restrict__ H, const float* __restrict__ dinv,
                              float* __restrict__ out, int n, int fshift) {
  int t = blockIdx.x * blockDim.x + threadIdx.x;
  int i = t >> fshift;
  if (i >= n) return;
  float di = dinv[i];
  out[t] = H[t] * di * di;
}

// One wave per edge; lane = feature. out[dst] += H[src] * dinv[src]*dinv[dst]
__global__ void edge_scatter(const int* __restrict__ src, const int* __restrict__ dst,
                             const float* __restrict__ dinv, const float* __restrict__ H,
                             float* __restrict__ out, int e, int F) {
  int t = blockIdx.x * blockDim.x + threadIdx.x;
  int ed = t >> 5;
  if (ed >= e) return;
  int lane = t & 31;
  __builtin_prefetch(src + ed + 4096, 0, 1);       // global_prefetch_b8 (speculative)
  __builtin_prefetch(dst + ed + 4096, 0, 1);
  int s = src[ed], d = dst[ed];
  float w = dinv[s] * dinv[d];
  const float* hr = H + (size_t)s * F;
  float* orow = out + (size_t)d * F;
  for (int f = lane; f < F; f += 32) atomicAdd(&orow[f], hr[f] * w);
}

__global__ void bias_act(float* __restrict__ X, const float* __restrict__ b,
                         int n, int fshift, int doRelu) {
  int t = blockIdx.x * blockDim.x + threadIdx.x;
  int i = t >> fshift;
  if (i >= n) return;
  int f = t & ((1 << fshift) - 1);
  float v = X[t] + b[f];
  X[t] = doRelu ? fmaxf(v, 0.0f) : v;
}

__global__ void zero_f32(float* __restrict__ p, int n) {
  int t = blockIdx.x * blockDim.x + threadIdx.x;
  if (t < n) p[t] = 0.0f;
}

// ---------------------------------------------------------------------------
// Attention pooling (F3 = 32). mean(x@W,0) == mean(x,0)@W (linearity).
// ---------------------------------------------------------------------------
__global__ void colsum32(const float* __restrict__ A, float* __restrict__ msum, int n) {
  int t = blockIdx.x * blockDim.x + threadIdx.x;
  int col = t & 31;
  int r0 = t >> 5;
  int step = (gridDim.x * blockDim.x) >> 5;
  float acc = 0.0f;
  for (int r = r0; r < n; r += step) acc += A[(size_t)r * 32 + col];
  atomicAdd(&msum[col], acc);
}

__global__ void ctx_kernel(const float* __restrict__ msum, const float* __restrict__ Watt,
                           float* __restrict__ ctx, float invN) {
  int j = threadIdx.x;                             // 0..31
  float s = 0.0f;
  for (int f = 0; f < 32; ++f) s += (msum[f] * invN) * Watt[f * 32 + j];
  ctx[j] = tanhf(s);
}

// wave per row: sc = sigmoid(a_i . ctx); p[f] += a[i,f]*sc
__global__ void att_pool(const float* __restrict__ A, const float* __restrict__ ctx,
                         float* __restrict__ p, int n) {
  int t = blockIdx.x * blockDim.x + threadIdx.x;
  int r = t >> 5;
  if (r >= n) return;
  int lane = t & 31;
  float a = A[(size_t)r * 32 + lane];
  float v = a * ctx[lane];
#pragma unroll
  for (int off = 16; off > 0; off >>= 1) v += __shfl_xor(v, off, 32);
  float sc = 1.0f / (1.0f + __expf(-v));
  atomicAdd(&p[lane], a * sc);
}

// ---------------------------------------------------------------------------
// NTN: out[k] = relu( p1^T Wtn[:,:,k] p2 + Wblk[k,:]@[p1;p2] + btn[k] )
// Wtn row-major (32,32,16): Wtn[i,j,k] = Wtn[(i*32+j)*16 + k]
// ---------------------------------------------------------------------------
__global__ void ntn_kernel(const float* __restrict__ p1, const float* __restrict__ p2,
                           const float* __restrict__ Wtn, const float* __restrict__ Wblk,
                           const float* __restrict__ btn, float* __restrict__ out) {
  int k = threadIdx.x;
  if (k >= 16) return;
  float s = 0.0f;
  for (int i = 0; i < 32; ++i) {
    float e1 = p1[i];
    for (int j = 0; j < 32; ++j) s += e1 * Wtn[(i * 32 + j) * 16 + k] * p2[j];
  }
  float blk = 0.0f;
  for (int f = 0; f < 32; ++f)
    blk += Wblk[k * 64 + f] * p1[f] + Wblk[k * 64 + 32 + f] * p2[f];
  out[k] = fmaxf(s + blk + btn[k], 0.0f);
}

// ---------------------------------------------------------------------------
extern "C" void kernel_launch(void* const* d_in, const int* in_sizes, int n_in,
                              void* d_out, int out_size, void* d_ws, size_t ws_size,
                              hipStream_t stream) {
  const float* X1   = (const float*)d_in[0];
  const int*   EI1  = (const int*)d_in[1];
  const float* X2   = (const float*)d_in[2];
  const int*   EI2  = (const int*)d_in[3];
  const float* W1   = (const float*)d_in[4];
  const float* b1   = (const float*)d_in[5];
  const float* W2   = (const float*)d_in[6];
  const float* b2   = (const float*)d_in[7];
  const float* W3   = (const float*)d_in[8];
  const float* b3   = (const float*)d_in[9];
  const float* Watt = (const float*)d_in[10];
  const float* Wtn  = (const float*)d_in[11];
  const float* Wblk = (const float*)d_in[12];
  const float* btn  = (const float*)d_in[13];
  (void)in_sizes; (void)n_in; (void)out_size; (void)ws_size;

  float* ws   = (float*)d_ws;
  float* B0   = ws;                                // N*64 (GEMM output / messages)
  float* B1   = B0 + (size_t)NNODES * 64;          // N*64 (aggregated features)
  float* deg  = B1 + (size_t)NNODES * 64;          // N    (becomes d^-1/2)
  float* msum = deg + NNODES;                      // 32
  float* ctx  = msum + 32;                         // 32
  float* p1   = ctx + 32;                          // 32
  float* p2   = p1 + 32;                           // 32

  auto cdiv = [](long a, long b) { return (int)((a + b - 1) / b); };
  const int gemmBlocks = cdiv((long)(NNODES / 16) * 32, 256);  // wave per m-tile

  auto run_graph = [&](const float* X, const int* ei, float* pout) {
    const int* src = ei;
    const int* dst = ei + NEDGES;

    deg_init<<<cdiv(NNODES, 256), 256, 0, stream>>>(deg, NNODES);
    deg_count<<<cdiv(NEDGES, 256), 256, 0, stream>>>(dst, deg, NEDGES);
    deg_fin<<<cdiv(NNODES, 256), 256, 0, stream>>>(deg, NNODES);

    auto layer = [&](const float* in, const float* Wm, const float* bv,
                     int Fout, int relu) {
      if (Fout == 64)
        gemm_wmma_lds<4><<<gemmBlocks, 256, 0, stream>>>(in, Wm, B0, NNODES, 64);
      else
        gemm_wmma_lds<2><<<gemmBlocks, 256, 0, stream>>>(in, Wm, B0, NNODES, 64);
      int fs = (Fout == 64) ? 6 : 5;
      selfloop_init<<<cdiv((long)NNODES * Fout, 256), 256, 0, stream>>>(
          B0, deg, B1, NNODES, fs);
      edge_scatter<<<cdiv((long)NEDGES * 32, 256), 256, 0, stream>>>(
          src, dst, deg, B0, B1, NEDGES, Fout);
      bias_act<<<cdiv((long)NNODES * Fout, 256), 256, 0, stream>>>(
          B1, bv, NNODES, fs, relu);
    };

    layer(X,  W1, b1, 64, 1);   // input buffer dead after GEMM -> B1 reuse is safe
    layer(B1, W2, b2, 64, 1);
    layer(B1, W3, b3, 32, 0);   // final features: B1[0 : N*32]

    zero_f32<<<1, 32, 0, stream>>>(msum, 32);
    zero_f32<<<1, 32, 0, stream>>>(pout, 32);
    colsum32<<<64, 256, 0, stream>>>(B1, msum, NNODES);
    ctx_kernel<<<1, 32, 0, stream>>>(msum, Watt, ctx, 1.0f / (float)NNODES);
    att_pool<<<cdiv((long)NNODES * 32, 256), 256, 0, stream>>>(B1, ctx, pout, NNODES);
  };

  run_graph(X1, EI1, p1);
  run_graph(X2, EI2, p2);
  ntn_kernel<<<1, 16, 0, stream>>>(p1, p2, Wtn, Wblk, btn, (float*)d_out);
}